// TResMF_71966472012154
// MI455X (gfx1250) — compile-verified
//
#include <hip/hip_runtime.h>

#define NUM_USERS 100000
#define NUM_ITEMS 50000
#define N_NODES   150000          // NUM_USERS + NUM_ITEMS
#define EMB_DIM   64
#define N_EDGES   2000000
#define N_LAYERS  3

typedef float v2f __attribute__((ext_vector_type(2)));
typedef float v8f __attribute__((ext_vector_type(8)));

// ---------------------------------------------------------------------------
// 1) init: ego = concat(user_emb, item_emb); acc = ego  (float4 vectorized)
//    total = 150000*64 = 9.6M floats = 2.4M float4 -> grid 9375 x 256 exact.
// ---------------------------------------------------------------------------
__global__ void lgcn_init(const float4* __restrict__ user,
                          const float4* __restrict__ item,
                          float4* __restrict__ ego,
                          float4* __restrict__ acc) {
    const size_t i  = (size_t)blockIdx.x * blockDim.x + threadIdx.x;
    const size_t ub = (size_t)NUM_USERS * EMB_DIM / 4;   // 1.6M, 4-aligned
    float4 v = (i < ub) ? user[i] : item[i - ub];
    ego[i] = v;
    acc[i] = v;
}

// ---------------------------------------------------------------------------
// 2) zero next-layer buffer (float4), grid 9375 x 256 exact.
// ---------------------------------------------------------------------------
__global__ void lgcn_zero(float4* __restrict__ p) {
    const size_t i = (size_t)blockIdx.x * blockDim.x + threadIdx.x;
    p[i] = make_float4(0.f, 0.f, 0.f, 0.f);
}

// ---------------------------------------------------------------------------
// 3) scatter: one wave32 per edge. 32 lanes x float2 = one 256B row gather of
//    ego[dst], scaled by edge_norm, scatter-added into nxt[src] with
//    fire-and-forget f32 atomics (L2-resident: 38.4MB buffer << 192MB L2).
//    Persistent waves + global_prefetch_b8 of next iteration's source row.
// ---------------------------------------------------------------------------
__global__ void lgcn_scatter(const float* __restrict__ ego,
                             const int*   __restrict__ src,
                             const int*   __restrict__ dst,
                             const float* __restrict__ norm,
                             float*       __restrict__ nxt) {
    const int       lane   = threadIdx.x & 31;
    const long long wave0  = ((long long)blockIdx.x * blockDim.x + threadIdx.x) >> 5;
    const long long stride = ((long long)gridDim.x * blockDim.x) >> 5;

    for (long long e = wave0; e < N_EDGES; e += stride) {
        // prefetch the row this wave will gather next iteration (gfx1250
        // global_prefetch_b8; speculative, pulls line into L2/WGP$)
        const long long en = e + stride;
        if (en < N_EDGES) {
            const int dp = dst[en];
            __builtin_prefetch(&ego[(size_t)dp * EMB_DIM + 2 * lane], 0, 3);
        }

        const int   d = dst[e];
        const int   s = src[e];
        const float w = norm[e];

        // coalesced 256B row load: 32 lanes x 8B
        const v2f v = *(const v2f*)&ego[(size_t)d * EMB_DIM + 2 * lane];

        float* p = &nxt[(size_t)s * EMB_DIM + 2 * lane];
        unsafeAtomicAdd(p,     v.x * w);   // global_atomic_add_f32, no return
        unsafeAtomicAdd(p + 1, v.y * w);
    }
}

// ---------------------------------------------------------------------------
// 4) acc += layer output (float4), grid 9375 x 256 exact.
// ---------------------------------------------------------------------------
__global__ void lgcn_accum(float4* __restrict__ acc, const float4* __restrict__ e) {
    const size_t i = (size_t)blockIdx.x * blockDim.x + threadIdx.x;
    float4 a = acc[i], b = e[i];
    a.x += b.x; a.y += b.y; a.z += b.z; a.w += b.w;
    acc[i] = a;
}

// ---------------------------------------------------------------------------
// 5) finalize: out = 0.25 * acc, in place, through the matrix pipe.
//    One wave per 16x16 tile.  D = sum_{p=0..3} A_p(16x4) x (0.25*E_p)(4x16)
//    == 0.25 * ACC(16x16), exact in f32 (V_WMMA_F32_16X16X4_F32).
//    Layouts per CDNA5 ISA 7.12.2:
//      A/B: K = vgpr + 2*(lane>=16), A row / B col = lane&15
//      C/D: M = vgpr + 8*(lane>=16), N = lane&15
//    37500 tiles = 3750 blocks x 10 waves: every wave owns a tile -> EXEC
//    is all-ones for the WMMA (ISA requirement).
// ---------------------------------------------------------------------------
__global__ void lgcn_finalize_wmma(float* __restrict__ data) {
    const int lane = threadIdx.x & 31;
    const int half = lane >> 4;        // 0: lanes 0-15, 1: lanes 16-31
    const int l15  = lane & 15;

    const int tilesPerRow = EMB_DIM / 16;                    // 4
    const long long tile  = (long long)blockIdx.x * (blockDim.x >> 5)
                          + (threadIdx.x >> 5);
    const int trow = (int)(tile / tilesPerRow);              // node block
    const int tcol = (int)(tile % tilesPerRow);              // dim block

    // A-matrix source row for this lane (row = lane&15 in both halves)
    const float* arow = data + (size_t)(trow * 16 + l15) * EMB_DIM + tcol * 16;

    v8f c = {};
#pragma unroll
    for (int p = 0; p < 4; ++p) {
        v2f a;                         // a[r] = ACC[row][4p + r + 2*half]
        const int col = 4 * p + 2 * half;
        a.x = arow[col];
        a.y = arow[col + 1];

        v2f b;                         // b[r] = 0.25 if N == 4p + K else 0
        b.x = (l15 == col + 0) ? 0.25f : 0.0f;
        b.y = (l15 == col + 1) ? 0.25f : 0.0f;

        c = __builtin_amdgcn_wmma_f32_16x16x4_f32(
                /*neg_a=*/false, a, /*neg_b=*/false, b,
                /*c_mod=*/(short)0, c, /*reuse_a=*/false, /*reuse_b=*/false);
    }

    // store D: element (r, lane) -> row M = r + 8*half, col N = l15
    float* out = data + (size_t)(trow * 16) * EMB_DIM + tcol * 16;
#pragma unroll
    for (int r = 0; r < 8; ++r)
        out[(size_t)(r + 8 * half) * EMB_DIM + l15] = c[r];
}

// ---------------------------------------------------------------------------
// host launcher
// ---------------------------------------------------------------------------
extern "C" void kernel_launch(void* const* d_in, const int* in_sizes, int n_in,
                              void* d_out, int out_size, void* d_ws, size_t ws_size,
                              hipStream_t stream) {
    const float* user = (const float*)d_in[0];   // [100000, 64]
    const float* item = (const float*)d_in[1];   // [50000, 64]
    const float* norm = (const float*)d_in[2];   // [2M]
    const int*   src  = (const int*)  d_in[3];   // [2M] int32
    const int*   dst  = (const int*)  d_in[4];   // [2M] int32

    float* acc  = (float*)d_out;                           // [150000*64] accumulator
    float* buf0 = (float*)d_ws;                            // ego (ping)
    float* buf1 = buf0 + (size_t)N_NODES * EMB_DIM;        // ego (pong)

    const dim3 blkV(256);
    const dim3 grdV((N_NODES * EMB_DIM / 4) / 256);        // 9375, exact

    lgcn_init<<<grdV, blkV, 0, stream>>>((const float4*)user, (const float4*)item,
                                         (float4*)buf0, (float4*)acc);

    float* cur = buf0;
    float* nxt = buf1;
    for (int l = 0; l < N_LAYERS; ++l) {
        lgcn_zero<<<grdV, blkV, 0, stream>>>((float4*)nxt);
        lgcn_scatter<<<4096, 256, 0, stream>>>(cur, src, dst, norm, nxt);
        lgcn_accum<<<grdV, blkV, 0, stream>>>((float4*)acc, (const float4*)nxt);
        float* t = cur; cur = nxt; nxt = t;
    }

    // 37500 tiles / 10 waves-per-block = 3750 blocks, 320 threads (10 wave32s)
    lgcn_finalize_wmma<<<3750, 320, 0, stream>>>(acc);
}